// TopologyPenalty_17970143167270
// MI455X (gfx1250) — compile-verified
//
#include <hip/hip_runtime.h>
#include <hip/hip_bf16.h>

// ---- constants from the reference ----
#define NRES     2048
#define SEQ_GAP  4
#define SIGMA    4.0f
#define EPS_NORM 1e-8f
#define EPS_DEN  1e-12f
#define CLASH_W  1.0f
#define WRITHE_W 0.5f
#define INV_4PI  0.07957747154594767f   // 1/(4*pi)

typedef float        v2f __attribute__((ext_vector_type(2)));
typedef float        v8f __attribute__((ext_vector_type(8)));
typedef unsigned int v4u __attribute__((ext_vector_type(4)));
typedef int          v4i __attribute__((ext_vector_type(4)));
typedef int          v8i __attribute__((ext_vector_type(8)));

// -------------------- utility --------------------

__global__ void init_acc_kernel(float* acc, int n) {
    int t = blockIdx.x * blockDim.x + threadIdx.x;
    if (t < n) acc[t] = 0.0f;
}

__device__ __forceinline__ float wave_reduce_add(float v) {
    #pragma unroll
    for (int off = 16; off > 0; off >>= 1)
        v += __shfl_down(v, off, 32);
    return v;
}

// -------------------- clash: WMMA f32 16x16x4 distance tiles --------------------
//
// d2(i,j) = |ri|^2 + |rj|^2 - 2 ri.rj
// A row m = (x,y,z,1); B col n = (-2x,-2y,-2z,|r|^2); C = |r_i|^2  =>  D = d2 tile.
// One wave sweeps JCHUNK j-tiles for a fixed i-tile: A and C operands are loaded
// once and reused across 8 WMMAs; only the B operand (3 floats/lane) is reloaded.
// Wave/tile indices are forced into SGPRs via readfirstlane so all control flow
// around the WMMAs is scalar -> EXEC provably stays all-ones at every v_wmma.

#define JCHUNK 8

__global__ __launch_bounds__(256) void clash_kernel(const float* __restrict__ coords,
                                                    float* __restrict__ clash_acc,
                                                    int B, int N) {
    // wave index is uniform across the wave: make it an SGPR value
    const int wave = __builtin_amdgcn_readfirstlane((int)(threadIdx.x >> 5));
    const int lane = threadIdx.x & 31;
    const int nt   = N >> 4;                   // j/i tiles per dim (128)
    const int nch  = nt / JCHUNK;              // j-chunks per row (16)
    const int wid  = blockIdx.x * 8 + wave;    // 8 waves per block, SGPR
    if (wid >= B * nt * nch) return;           // scalar branch
    const int b    = wid / (nt * nch);
    const int rem  = wid - b * (nt * nch);
    const int ti   = rem / nch;
    const int ch   = rem - ti * nch;
    if (ch * JCHUNK + (JCHUNK - 1) < ti) return;   // entire chunk below diagonal; scalar

    const float* c = coords + (size_t)b * N * 3;
    const int i0   = ti << 4;
    const int half = lane >> 4;                // 0: lanes 0-15, 1: lanes 16-31
    const int l15  = lane & 15;

    // A operand (i-tile invariant): row m = i0+l15 as (x,y,z,1)
    const int m = i0 + l15;
    const float amx = c[3*m+0], amy = c[3*m+1], amz = c[3*m+2];
    v2f A;
    A.x = half ? amz  : amx;    // K2 : K0
    A.y = half ? 1.0f : amy;    // K3 : K1

    // C operand (i-tile invariant): |r_i|^2 per D slot
    v8f C;
    #pragma unroll
    for (int v = 0; v < 8; ++v) {
        const int mr = i0 + v + (half << 3);
        const float x = c[3*mr+0], y = c[3*mr+1], z = c[3*mr+2];
        C[v] = x*x + y*y + z*z;
    }

    const float kExp = -1.0f / (2.0f * SIGMA * SIGMA);
    float s = 0.0f;

    #pragma unroll
    for (int u = 0; u < JCHUNK; ++u) {
        const int tj = ch * JCHUNK + u;        // SGPR: scalar branch below
        if (tj >= ti) {
            const int j0 = tj << 4;
            const int n  = j0 + l15;
            const float bnx = c[3*n+0], bny = c[3*n+1], bnz = c[3*n+2];
            const float nrm_j = bnx*bnx + bny*bny + bnz*bnz;
            v2f Bm;
            Bm.x = half ? (-2.0f*bnz) : (-2.0f*bnx);
            Bm.y = half ? nrm_j       : (-2.0f*bny);

            v8f D = __builtin_amdgcn_wmma_f32_16x16x4_f32(
                false, A, false, Bm, (short)0, C, false, false);

            const int j = j0 + l15;
            #pragma unroll
            for (int v = 0; v < 8; ++v) {
                const int i = i0 + v + (half << 3);
                const float g = __expf(D[v] * kExp);
                s += ((j - i) > SEQ_GAP) ? g : 0.0f;
            }
        }
    }

    s = wave_reduce_add(s);
    if (lane == 0) atomicAdd(&clash_acc[b], s);
}

// -------------------- TDM: 1-D tensor load into LDS --------------------
//
// D# per CDNA5 ISA ch.8: group0 = {count=1, lds_addr, global_addr, type=2},
// group1 = {data_size=4B, tensor_dim0=tile_dim0=stride0=nelem, tensor_dim1=1},
// groups 2/3/4 zero (<=2D tensor). One wave issues; TENSORcnt tracks completion.
// This toolchain's builtin is the 6-arg clang-23 form:
//   (uint32x4 g0, int32x8 g1, int32x4 g2, int32x4 g3, int32x8 g4, i32 cpol)

__device__ __forceinline__ void tdm_load_f32_1d(unsigned int lds_byte,
                                                const float* gptr,
                                                unsigned int nelem) {
#if __has_builtin(__builtin_amdgcn_tensor_load_to_lds)
    const unsigned long long ga = (unsigned long long)(uintptr_t)gptr;
    v4u g0;
    g0[0] = 1u;                                           // count=1, is_restore=0
    g0[1] = lds_byte;                                     // lds_addr [63:32]
    g0[2] = (unsigned int)(ga & 0xFFFFFFFFu);             // global_addr [95:64]
    g0[3] = (unsigned int)((ga >> 32) & 0x01FFFFFFu)      // global_addr [120:96]
          | (2u << 30);                                   // type=2 (image)
    v8i g1;
    g1[0] = (int)(2u << 16);                              // data_size=2 -> 4 bytes
    g1[1] = (int)((nelem & 0xFFFFu) << 16);               // tensor_dim0[15:0]  @48
    g1[2] = (int)(((nelem >> 16) & 0xFFFFu)               // tensor_dim0[31:16] @64
          | (1u << 16));                                  // tensor_dim1 = 1    @80
    g1[3] = (int)((nelem & 0xFFFFu) << 16);               // tile_dim0 @112
    g1[4] = 0;                                            // tile_dim1/2 = 0 (1-D)
    g1[5] = (int)nelem;                                   // tensor_dim0_stride[31:0]
    g1[6] = 0;
    g1[7] = 0;
    v4i g2 = {0, 0, 0, 0};
    v4i g3 = {0, 0, 0, 0};
    v8i g4 = {0, 0, 0, 0, 0, 0, 0, 0};
    __builtin_amdgcn_tensor_load_to_lds(g0, g1, g2, g3, g4, 0);
#else
    (void)lds_byte; (void)gptr; (void)nelem;
#endif
}

// -------------------- writhe: LDS-tiled per-pair solid angles --------------------

struct F3 { float x, y, z; };

__device__ __forceinline__ F3 f3lds(const float* p, int i) {
    F3 r; r.x = p[3*i+0]; r.y = p[3*i+1]; r.z = p[3*i+2]; return r;
}
__device__ __forceinline__ F3 f3sub(F3 a, F3 b) {
    F3 r; r.x = a.x-b.x; r.y = a.y-b.y; r.z = a.z-b.z; return r;
}
__device__ __forceinline__ float f3dot(F3 a, F3 b) {
    return a.x*b.x + a.y*b.y + a.z*b.z;
}
__device__ __forceinline__ F3 f3cross(F3 a, F3 b) {
    F3 r;
    r.x = a.y*b.z - a.z*b.y;
    r.y = a.z*b.x - a.x*b.z;
    r.z = a.x*b.y - a.y*b.x;
    return r;
}
__device__ __forceinline__ F3 f3unit(F3 r, float* nrm) {
    const float n = sqrtf(f3dot(r, r));
    *nrm = n;
    const float inv = 1.0f / fmaxf(n, EPS_NORM);
    F3 u; u.x = r.x*inv; u.y = r.y*inv; u.z = r.z*inv; return u;
}
__device__ __forceinline__ float tri_solid_angle(F3 v1, F3 v2, F3 v3) {
    const float cnum = f3dot(v1, f3cross(v2, v3));
    const float den  = 1.0f + f3dot(v1, v2) + f3dot(v2, v3) + f3dot(v1, v3);
    return (fabsf(den) < EPS_DEN) ? 0.0f : 2.0f * atan2f(cnum, den);
}

#define WT_TI 16
#define WT_TJ 256

__global__ __launch_bounds__(256) void writhe_kernel(const float* __restrict__ coords,
                                                     float* __restrict__ wr_acc,
                                                     int B, int N) {
    __shared__ float ldsI[(WT_TI + 1) * 3];
    __shared__ float ldsJ[(WT_TJ + 1) * 3];

    const int S  = N - 1;
    const int b  = blockIdx.z;
    const int i0 = blockIdx.y * WT_TI;
    const int j0 = blockIdx.x * WT_TJ;
    // whole tile masked (max j - min i <= SEQ_GAP)?  block-uniform early out
    if (j0 + WT_TJ - 1 <= i0 + SEQ_GAP) return;

    const float* c = coords + (size_t)b * N * 3;
    const int nPtsI = min(WT_TI + 1, N - i0);
    const int nPtsJ = min(WT_TJ + 1, N - j0);

#if __has_builtin(__builtin_amdgcn_tensor_load_to_lds) && __has_builtin(__builtin_amdgcn_s_wait_tensorcnt)
    {
        const int wv = __builtin_amdgcn_readfirstlane((int)(threadIdx.x >> 5));
        if (wv == 0) {           // one wave issues both TDM descriptors (scalar branch)
            tdm_load_f32_1d((unsigned int)(uintptr_t)&ldsI[0], c + (size_t)i0 * 3,
                            (unsigned int)(nPtsI * 3));
            tdm_load_f32_1d((unsigned int)(uintptr_t)&ldsJ[0], c + (size_t)j0 * 3,
                            (unsigned int)(nPtsJ * 3));
            __builtin_amdgcn_s_wait_tensorcnt(0);
        }
    }
    // compiler must not assume LDS is unwritten (TDM wrote it behind its back)
    asm volatile("" ::: "memory");
#else
    for (int t = threadIdx.x; t < nPtsI * 3; t += blockDim.x) ldsI[t] = c[i0 * 3 + t];
    for (int t = threadIdx.x; t < nPtsJ * 3; t += blockDim.x) ldsJ[t] = c[j0 * 3 + t];
#endif
    __syncthreads();

    const int li  = threadIdx.x >> 4;          // 0..15  (i-row within tile)
    const int lj0 = threadIdx.x & 15;          // j lane offset
    const int i   = i0 + li;

    float s = 0.0f;
    if (i < S) {
        const F3 a1 = f3lds(ldsI, li);
        const F3 a2 = f3lds(ldsI, li + 1);
        #pragma unroll
        for (int k = 0; k < WT_TJ / 16; ++k) {
            const int lj = lj0 + 16 * k;
            const int j  = j0 + lj;
            if (j < S && (j - i) > SEQ_GAP) {
                const F3 b1 = f3lds(ldsJ, lj);
                const F3 b2 = f3lds(ldsJ, lj + 1);
                float n11, n12, n21, n22;
                const F3 u11 = f3unit(f3sub(a1, b1), &n11);
                const F3 u12 = f3unit(f3sub(a1, b2), &n12);
                const F3 u21 = f3unit(f3sub(a2, b1), &n21);
                const F3 u22 = f3unit(f3sub(a2, b2), &n22);
                if ((n11 >= EPS_NORM) && (n12 >= EPS_NORM) &&
                    (n21 >= EPS_NORM) && (n22 >= EPS_NORM)) {
                    s += tri_solid_angle(u11, u21, u22) + tri_solid_angle(u11, u22, u12);
                }
            }
        }
    }
    s = wave_reduce_add(s);
    if ((threadIdx.x & 31) == 0 && s != 0.0f) atomicAdd(&wr_acc[b], s);
}

// -------------------- final combine --------------------

__global__ void finish_kernel(const float* __restrict__ acc, float* __restrict__ out, int B) {
    float clash = 0.0f, wr2 = 0.0f;
    for (int b = 0; b < B; ++b) {
        clash += acc[b];
        const float w = acc[B + b] * INV_4PI;
        wr2 += w * w;
    }
    out[0] = CLASH_W * (clash / (float)B) + WRITHE_W * (wr2 / (float)B);
}

// -------------------- host launcher --------------------

extern "C" void kernel_launch(void* const* d_in, const int* in_sizes, int n_in,
                              void* d_out, int out_size, void* d_ws, size_t ws_size,
                              hipStream_t stream) {
    const float* coords = (const float*)d_in[0];
    float* out = (float*)d_out;

    const int N = NRES;
    const int B = in_sizes[0] / (3 * N);
    float* acc = (float*)d_ws;   // [0..B) clash, [B..2B) writhe-sum

    init_acc_kernel<<<1, 64, 0, stream>>>(acc, 2 * B);

    // clash: one wave per (i-tile, 8-j-tile chunk), 8 waves per block
    const int nt  = N >> 4;
    const int nch = nt / JCHUNK;
    const int clashWaves  = B * nt * nch;
    const int clashBlocks = (clashWaves + 7) / 8;
    clash_kernel<<<clashBlocks, 256, 0, stream>>>(coords, acc, B, N);

    // writhe: 16x256 pair tiles staged through LDS (TDM)
    const int S = N - 1;
    dim3 wgrid((S + WT_TJ - 1) / WT_TJ, (S + WT_TI - 1) / WT_TI, B);
    writhe_kernel<<<wgrid, 256, 0, stream>>>(coords, acc + B, B, N);

    finish_kernel<<<1, 1, 0, stream>>>(acc, out, B);
}